// LogicLayer_54528904790310
// MI455X (gfx1250) — compile-verified
//
#include <hip/hip_runtime.h>
#include <hip/hip_bf16.h>

#define BATCH   4096
#define IN_DIM  8192
#define OUT_DIM 16384
#define ROWS    2          // batch rows staged in LDS per block (2 * 32KB = 64KB)

#define AS1 __attribute__((address_space(1)))
#define AS3 __attribute__((address_space(3)))

typedef float v2f __attribute__((ext_vector_type(2)));
typedef float v8f __attribute__((ext_vector_type(8)));
typedef int   v4i __attribute__((ext_vector_type(4)));

// Pack one gate's 4 affine coefficients (each in [-2,2]) into 3-bit fields.
__host__ __device__ constexpr unsigned packGate(int c0, int c1, int c2, int c3) {
    return (unsigned)(c0 + 2) | ((unsigned)(c1 + 2) << 3) |
           ((unsigned)(c2 + 2) << 6) | ((unsigned)(c3 + 2) << 9);
}
// GATE_COEFS rows (const, a, b, ab) for the 16 logic gates.
__device__ constexpr unsigned PK[16] = {
    packGate(0, 0, 0, 0),  packGate(0, 0, 0, 1),
    packGate(0, 1, 0, -1), packGate(0, 1, 0, 0),
    packGate(0, 0, 1, -1), packGate(0, 0, 1, 0),
    packGate(0, 1, 1, -2), packGate(0, 1, 1, -1),
    packGate(1, -1, -1, 1), packGate(1, -1, -1, 2),
    packGate(1, 0, -1, 0), packGate(1, 0, -1, 1),
    packGate(1, -1, 0, 0), packGate(1, -1, 0, 1),
    packGate(1, 0, 0, -1), packGate(1, 0, 0, 0),
};

// unpack coefficient n of packed gate word pk as float
__device__ inline float unpackCoef(unsigned pk, int n) {
    return (float)((int)((pk >> (3 * n)) & 7u) - 2);
}

// ---------------------------------------------------------------------------
// Kernel 1: coef[o][0..3] = softmax(weights[o][0..15]) @ GATE_COEFS  (fp32 WMMA)
// One wave32 handles 16 output neurons. A = P (16x16 probs, K split into 4
// chunks of K=4), B = GATE_COEFS padded to 16 columns (only N<4 nonzero).
// ---------------------------------------------------------------------------
__global__ __launch_bounds__(128) void LogicLayer_coef_kernel(
    const float* __restrict__ W, float* __restrict__ coef) {
    const int lane = threadIdx.x & 31;
    const int wave = (blockIdx.x * blockDim.x + threadIdx.x) >> 5;
    const int base = wave * 16;            // first neuron of this wave's tile
    const int r    = lane & 15;            // matrix row / column index 0..15
    const int half = lane >> 4;            // lane half selects K sub-pairs

    // per-lane softmax over the 16 gate logits of neuron (base + r)
    const float4* w4 = (const float4*)(W + (size_t)(base + r) * 16);
    float p[16];
    {
        const float4 l0 = w4[0], l1 = w4[1], l2 = w4[2], l3 = w4[3];
        p[0] = l0.x;  p[1] = l0.y;  p[2]  = l0.z;  p[3]  = l0.w;
        p[4] = l1.x;  p[5] = l1.y;  p[6]  = l1.z;  p[7]  = l1.w;
        p[8] = l2.x;  p[9] = l2.y;  p[10] = l2.z;  p[11] = l2.w;
        p[12] = l3.x; p[13] = l3.y; p[14] = l3.z;  p[15] = l3.w;
    }
    float m = -3.0e38f;
#pragma unroll
    for (int i = 0; i < 16; ++i) m = fmaxf(m, p[i]);
    float s = 0.f;
#pragma unroll
    for (int i = 0; i < 16; ++i) { p[i] = __expf(p[i] - m); s += p[i]; }
    const float inv = 1.0f / s;
#pragma unroll
    for (int i = 0; i < 16; ++i) p[i] *= inv;

#if __has_builtin(__builtin_amdgcn_wmma_f32_16x16x4_f32)
    const bool nlt4 = (r < 4);
    const int  sh   = 3 * r;               // shift for unpack when r < 4
    v8f c = {};
#pragma unroll
    for (int k0 = 0; k0 < 16; k0 += 4) {
        // A tile 16x4 f32: lanes 0-15 hold K={k0,k0+1}, lanes 16-31 K={k0+2,k0+3}.
        // Select on VALUES (v_cndmask), never on addresses.
        const float q0 = p[k0 + 0], q1 = p[k0 + 1];
        const float q2 = p[k0 + 2], q3 = p[k0 + 3];
        v2f a;
        a.x = half ? q2 : q0;
        a.y = half ? q3 : q1;
        // B tile 4x16 f32: rows K striped like A, columns N = r (pad N>=4 with 0).
        // Packed gate words are compile-time constants; 'half' picks between two
        // immediates, then a shift/mask unpacks component r.
        const unsigned pb0 = half ? PK[k0 + 2] : PK[k0 + 0];
        const unsigned pb1 = half ? PK[k0 + 3] : PK[k0 + 1];
        v2f b;
        b.x = nlt4 ? (float)((int)((pb0 >> sh) & 7u) - 2) : 0.f;
        b.y = nlt4 ? (float)((int)((pb1 >> sh) & 7u) - 2) : 0.f;
        c = __builtin_amdgcn_wmma_f32_16x16x4_f32(false, a, false, b,
                                                  (short)0, c, false, false);
    }
    // D layout: VGPR j holds M = j + 8*half, N = r. Only N<4 is a real coef.
    if (nlt4) {
#pragma unroll
        for (int j = 0; j < 8; ++j) {
            const int Mrow = j + 8 * half;
            coef[(size_t)(base + Mrow) * 4 + r] = c[j];
        }
    }
#else
    // scalar fallback (no WMMA builtin available)
    if (half == 0) {
        float acc[4] = {0.f, 0.f, 0.f, 0.f};
#pragma unroll
        for (int k = 0; k < 16; ++k) {
#pragma unroll
            for (int n = 0; n < 4; ++n) acc[n] += p[k] * unpackCoef(PK[k], n);
        }
#pragma unroll
        for (int n = 0; n < 4; ++n) coef[(size_t)(base + r) * 4 + n] = acc[n];
    }
#endif
}

// ---------------------------------------------------------------------------
// Kernel 2: stage ROWS rows of x into LDS via async global->LDS DMA, then
// serve the random column gathers from LDS banks; non-temporal output stores.
// ---------------------------------------------------------------------------
__global__ __launch_bounds__(256) void LogicLayer_gather_kernel(
    const float* __restrict__ x, const float* __restrict__ coef,
    const int* __restrict__ idx_a, const int* __restrict__ idx_b,
    float* __restrict__ out) {
    __shared__ float rows[ROWS][IN_DIM];   // 64 KB

    const int t  = threadIdx.x;
    const int b0 = blockIdx.x * ROWS;

#if __has_builtin(__builtin_amdgcn_global_load_async_to_lds_b128)
    {
        AS1 char* gsrc  = (AS1 char*)(x + (size_t)b0 * IN_DIM);
        AS3 char* lbase = (AS3 char*)(&rows[0][0]);
        // ROWS*IN_DIM*4 bytes / (256 threads * 16B) = 16 async b128 per thread
#pragma unroll
        for (int i = 0; i < (ROWS * IN_DIM * 4) / (256 * 16); ++i) {
            const int off = (i * 256 + t) * 16;
            __builtin_amdgcn_global_load_async_to_lds_b128(
                (AS1 v4i*)(gsrc + off), (AS3 v4i*)(lbase + off), 0, 0);
        }
#if __has_builtin(__builtin_amdgcn_s_wait_asynccnt)
        __builtin_amdgcn_s_wait_asynccnt(0);
#else
        asm volatile("s_wait_asynccnt 0" ::: "memory");
#endif
    }
#else
    {
        const float4* src = (const float4*)(x + (size_t)b0 * IN_DIM);
        float4*       dst = (float4*)(&rows[0][0]);
        for (int i = t; i < ROWS * IN_DIM / 4; i += 256) dst[i] = src[i];
    }
#endif
    __syncthreads();

    // 16384 cols / 256 threads = 64 iterations; coalesced idx/coef loads and
    // coalesced NT stores; gathers hit LDS (64 banks per WGP half).
    for (int o = t; o < OUT_DIM; o += 256) {
        const int    iA = idx_a[o];
        const int    iB = idx_b[o];
        const float4 cf = ((const float4*)coef)[o];
#pragma unroll
        for (int rr = 0; rr < ROWS; ++rr) {
            const float av = rows[rr][iA];
            const float bv = rows[rr][iB];
            const float v  = cf.x + cf.y * av + cf.z * bv + cf.w * (av * bv);
            __builtin_nontemporal_store(v, &out[(size_t)(b0 + rr) * OUT_DIM + o]);
        }
    }
}

extern "C" void kernel_launch(void* const* d_in, const int* in_sizes, int n_in,
                              void* d_out, int out_size, void* d_ws, size_t ws_size,
                              hipStream_t stream) {
    const float* x  = (const float*)d_in[0];
    const float* W  = (const float*)d_in[1];
    const int*   ia = (const int*)d_in[2];
    const int*   ib = (const int*)d_in[3];
    float*       out  = (float*)d_out;
    float*       coef = (float*)d_ws;      // OUT_DIM * 4 floats = 256 KB scratch

    // 1024 waves of 16 neurons each; 4 waves per block -> 256 blocks
    LogicLayer_coef_kernel<<<(OUT_DIM / 16) / 4, 128, 0, stream>>>(W, coef);
    // one block per 2 batch rows
    LogicLayer_gather_kernel<<<BATCH / ROWS, 256, 0, stream>>>(x, coef, ia, ib, out);
}